// DotLinkPredictor_10024453669138
// MI455X (gfx1250) — compile-verified
//
#include <hip/hip_runtime.h>

typedef __attribute__((ext_vector_type(2))) float v2f;
typedef __attribute__((ext_vector_type(8))) float v8f;

// D_FEAT = 64 (fixed by the reference).
// One wave (32 lanes) computes 16 edge dot-products per group via
// 16 chained V_WMMA_F32_16X16X4_F32 (K = 16 * 4 = 64), exact fp32 FMA math.
__global__ __launch_bounds__(256) void edge_dot_wmma_f32(
    const float* __restrict__ h,
    const int*   __restrict__ src,
    const int*   __restrict__ dst,
    float*       __restrict__ out,
    int n_edges)
{
    const int lane          = threadIdx.x & 31;
    const int waveInBlock   = threadIdx.x >> 5;
    const int wavesPerBlock = blockDim.x >> 5;
    const int waveGlobal    = blockIdx.x * wavesPerBlock + waveInBlock;
    const int nWavesTotal   = gridDim.x * wavesPerBlock;
    const int nGroups       = (n_edges + 15) >> 4;

    const int m = lane & 15;   // which edge of the 16 this lane serves (A rows / B cols)
    const int g = lane >> 4;   // half-wave selects K sub-offset {0,1} -> elements 2g, 2g+1

    for (int grp = waveGlobal; grp < nGroups; grp += nWavesTotal) {
        const int ebase = grp << 4;
        int em = ebase + m;
        if (em >= n_edges) em = n_edges - 1;   // clamp loads only; EXEC stays all-1s for WMMA

        const int si = src[em];
        const int di = dst[em];
        const float* __restrict__ pu = h + (long long)si * 64 + (g << 1);
        const float* __restrict__ pv = h + (long long)di * 64 + (g << 1);

        v8f acc = {};  // 16x16 f32 accumulator (8 VGPRs)
#pragma unroll
        for (int k = 0; k < 16; ++k) {
            // A chunk: A[M=m][K=4k+2g+{0,1}]  (lane L<16 <-> L+16 share a row, different K)
            v2f a = *(const v2f*)(pu + 4 * k);
            // B chunk: B[K=4k+2g+{0,1}][N=m] = hv[e_m][...] — same addressing, dst row
            v2f b = *(const v2f*)(pv + 4 * k);
            // 8-arg form: (neg_a, A, neg_b, B, c_mod, C, reuse_a, reuse_b)
            acc = __builtin_amdgcn_wmma_f32_16x16x4_f32(
                false, a, false, b, (short)0, acc, false, false);
        }

        // Diagonal D[m][m]:
        //   m in [0,8):  lane = m,      vgpr = m
        //   m in [8,16): lane = m + 16, vgpr = m - 8   (lanes 24..31 hold acc[lane-24])
        int sel = -1, edge = -1;
        if (lane < 8)        { sel = lane;      edge = ebase + lane; }
        else if (lane >= 24) { sel = lane - 24; edge = ebase + lane - 16; }

        if (sel >= 0 && edge < n_edges) {
            float v = acc[0];
#pragma unroll
            for (int i = 1; i < 8; ++i) v = (sel == i) ? acc[i] : v;
            out[edge] = v;
        }
    }
}

extern "C" void kernel_launch(void* const* d_in, const int* in_sizes, int n_in,
                              void* d_out, int out_size, void* d_ws, size_t ws_size,
                              hipStream_t stream) {
    const float* h   = (const float*)d_in[0];
    const int*   src = (const int*)d_in[1];
    const int*   dst = (const int*)d_in[2];
    float*       out = (float*)d_out;

    const int n_edges = in_sizes[1];          // 1,000,000
    const int nGroups = (n_edges + 15) / 16;  // 62,500 wave-groups

    const int threads       = 256;            // 8 waves per block (wave32)
    const int wavesPerBlock = threads / 32;
    int blocks = (nGroups + wavesPerBlock - 1) / wavesPerBlock;
    if (blocks < 1) blocks = 1;

    edge_dot_wmma_f32<<<blocks, threads, 0, stream>>>(h, src, dst, out, n_edges);
}